// GIN_68109591380381
// MI455X (gfx1250) — compile-verified
//
#include <hip/hip_runtime.h>
#include <math.h>

#define D_FEAT   1433
#define HIDDEN   16
#define N_CLASSES 7

typedef __attribute__((ext_vector_type(2))) float v2f;
typedef __attribute__((ext_vector_type(8))) float v8f;

// ---------------------------------------------------------------------------
// Kernel 0: zero the aggregation buffer (must re-zero every call: atomics
// accumulate into it and the harness does not re-poison between replays).
// ---------------------------------------------------------------------------
__global__ void gin_zero(float* __restrict__ p, int n) {
    int i = blockIdx.x * blockDim.x + threadIdx.x;
    if (i < n) p[i] = 0.0f;
}

// ---------------------------------------------------------------------------
// Kernel 1: y[N,16] = x[N,1433] @ w1[1433,16] via V_WMMA_F32_16X16X4_F32.
// One wave computes a full 16-row x 16-col tile; K-loop in steps of 4.
//
// A (16x4 f32) lane layout (ISA 7.12.2): lanes 0-15 hold K={k,k+1} of row M=lane,
// lanes 16-31 hold K={k+2,k+3} of row M=lane-16  -> each lane: one b64 load of x.
// B (4x16 f32): VGPR0 = row K=k (lanes 0-15) / K=k+2 (lanes 16-31), VGPR1 = K=k+1 / k+3.
// C/D (16x16 f32): VGPR v, lanes 0-15 -> M=v, lanes 16-31 -> M=8+v.
// ---------------------------------------------------------------------------
__global__ __launch_bounds__(256) void gin_gemm1(const float* __restrict__ x,
                                                 const float* __restrict__ w1,
                                                 float* __restrict__ y,
                                                 int n_nodes) {
    const int lane  = threadIdx.x & 31;
    const int wave  = threadIdx.x >> 5;
    const int tile  = blockIdx.x * (blockDim.x >> 5) + wave;
    const int row_base = tile * 16;
    if (row_base >= n_nodes) return;

    const int m    = lane & 15;          // A row within tile / B column
    const int koff = (lane >> 4) * 2;    // 0 for lanes 0-15, 2 for lanes 16-31

    const float* __restrict__ xrow = x + (size_t)(row_base + m) * D_FEAT;

    v8f c = {};
    const int kfull = (D_FEAT / 4) * 4;  // 1432
    int k = 0;
    #pragma unroll 4
    for (; k < kfull; k += 4) {
        // A: two consecutive K elements of this lane's row -> single b64 load
        v2f a = *(const v2f*)(xrow + k + koff);
        // B: w1 rows (k+koff) and (k+koff+1), column m (w1 is hot in L2/WGP$)
        v2f b;
        b.x = w1[(k + koff)     * HIDDEN + m];
        b.y = w1[(k + koff + 1) * HIDDEN + m];
        c = __builtin_amdgcn_wmma_f32_16x16x4_f32(false, a, false, b,
                                                  (short)0, c, false, false);
    }
    { // K tail (1433 % 4 == 1): zero-pad out-of-range K columns/rows
        v2f a, b;
        a.x = (k + koff + 0 < D_FEAT) ? xrow[k + koff + 0] : 0.0f;
        a.y = (k + koff + 1 < D_FEAT) ? xrow[k + koff + 1] : 0.0f;
        b.x = (k + koff + 0 < D_FEAT) ? w1[(k + koff + 0) * HIDDEN + m] : 0.0f;
        b.y = (k + koff + 1 < D_FEAT) ? w1[(k + koff + 1) * HIDDEN + m] : 0.0f;
        c = __builtin_amdgcn_wmma_f32_16x16x4_f32(false, a, false, b,
                                                  (short)0, c, false, false);
    }

    // Store D tile: row = row_base + (lane>=16 ? 8 : 0) + v, col = m
    const int rofs = (lane >> 4) * 8;
    #pragma unroll
    for (int v = 0; v < 8; ++v) {
        int row = row_base + rofs + v;
        if (row < n_nodes) y[(size_t)row * HIDDEN + m] = c[v];
    }
}

// ---------------------------------------------------------------------------
// Kernel 2: agg[dst] += y[src]  (per (edge, feature) thread; 16 consecutive
// threads cover one edge's 16-dim row -> coalesced gather, L2-side f32 atomics)
// ---------------------------------------------------------------------------
__global__ void gin_aggregate(const float* __restrict__ y,
                              const int* __restrict__ src,
                              const int* __restrict__ dst,
                              float* __restrict__ agg,
                              int n_edges) {
    int idx = blockIdx.x * blockDim.x + threadIdx.x;
    int e = idx >> 4;
    int f = idx & 15;
    if (e < n_edges) {
        int s = src[e];
        int d = dst[e];
        float v = y[(size_t)s * HIDDEN + f];
        atomicAdd(&agg[(size_t)d * HIDDEN + f], v);
    }
}

// ---------------------------------------------------------------------------
// Kernel 3: out = sigmoid( relu((1+eps)*y + agg + b1) @ w2 + b2 )
// One thread per node; 16x7 second layer is tiny -> VALU FMA chain.
// ---------------------------------------------------------------------------
__global__ void gin_finalize(const float* __restrict__ y,
                             const float* __restrict__ agg,
                             const float* __restrict__ b1,
                             const float* __restrict__ w2,
                             const float* __restrict__ b2,
                             const float* __restrict__ eps_p,
                             float* __restrict__ out,
                             int n_nodes) {
    int i = blockIdx.x * blockDim.x + threadIdx.x;
    if (i >= n_nodes) return;
    const float ep = 1.0f + eps_p[0];

    float h[HIDDEN];
    const float4* y4 = (const float4*)(y   + (size_t)i * HIDDEN);
    const float4* a4 = (const float4*)(agg + (size_t)i * HIDDEN);
    #pragma unroll
    for (int q = 0; q < 4; ++q) {
        float4 yv = y4[q];
        float4 av = a4[q];
        h[4*q+0] = fmaxf(fmaf(ep, yv.x, av.x) + b1[4*q+0], 0.0f);
        h[4*q+1] = fmaxf(fmaf(ep, yv.y, av.y) + b1[4*q+1], 0.0f);
        h[4*q+2] = fmaxf(fmaf(ep, yv.z, av.z) + b1[4*q+2], 0.0f);
        h[4*q+3] = fmaxf(fmaf(ep, yv.w, av.w) + b1[4*q+3], 0.0f);
    }

    #pragma unroll
    for (int cc = 0; cc < N_CLASSES; ++cc) {
        float o = b2[cc];
        #pragma unroll
        for (int j = 0; j < HIDDEN; ++j)
            o = fmaf(h[j], w2[j * N_CLASSES + cc], o);
        out[(size_t)i * N_CLASSES + cc] = 1.0f / (1.0f + __expf(-o));
    }
}

// ---------------------------------------------------------------------------
extern "C" void kernel_launch(void* const* d_in, const int* in_sizes, int n_in,
                              void* d_out, int out_size, void* d_ws, size_t ws_size,
                              hipStream_t stream) {
    const float* x    = (const float*)d_in[0];
    const int*   ei   = (const int*)  d_in[1];   // [2, n_edges] (src row, dst row)
    const float* w1   = (const float*)d_in[2];
    const float* b1   = (const float*)d_in[3];
    const float* w2   = (const float*)d_in[4];
    const float* b2   = (const float*)d_in[5];
    const float* eps  = (const float*)d_in[6];
    float*       out  = (float*)d_out;

    const int n_nodes = in_sizes[0] / D_FEAT;
    const int n_edges = in_sizes[1] / 2;

    float* y   = (float*)d_ws;                       // [n_nodes, 16]
    float* agg = y + (size_t)n_nodes * HIDDEN;       // [n_nodes, 16]
    const int* src = ei;
    const int* dst = ei + n_edges;

    // 0) zero agg (atomic accumulator)
    {
        int zn = n_nodes * HIDDEN;
        gin_zero<<<(zn + 255) / 256, 256, 0, stream>>>(agg, zn);
    }
    // 1) y = x @ w1   (WMMA, 8 waves/block, one 16x16 tile per wave)
    {
        int tiles = (n_nodes + 15) / 16;
        int blocks = (tiles + 7) / 8;
        gin_gemm1<<<blocks, 256, 0, stream>>>(x, w1, y, n_nodes);
    }
    // 2) agg[dst] += y[src]
    {
        long long total = (long long)n_edges * HIDDEN;
        int blocks = (int)((total + 255) / 256);
        gin_aggregate<<<blocks, 256, 0, stream>>>(y, src, dst, agg, n_edges);
    }
    // 3) fused epilogue
    {
        gin_finalize<<<(n_nodes + 255) / 256, 256, 0, stream>>>(
            y, agg, b1, w2, b2, eps, out, n_nodes);
    }
}